// DigitalPhaser_19155554140783
// MI455X (gfx1250) — compile-verified
//
#include <hip/hip_runtime.h>

// Digital phaser: time-varying linear IIR, solved as a chunked parallel scan.
//   state s = (h1, x1, h2, h3, h4)  (5-dim, padded to 8 for WMMA tiles)
//   s_t = M(p_t) s_{t-1} + c(p_t) x_t      -- affine, so chunks compose.
// Pass1: per chunk, zero-state response Z_i (64x8, batch-major) + transfer P_i (8x8)
// Pass2: sequential chunk scan  S <- P_i S + Z_i  using V_WMMA_F32_16X16X4_F32
// Pass3: per chunk, replay from correct initial state, write y = h4 + x.

typedef float v2f __attribute__((ext_vector_type(2)));
typedef float v8f __attribute__((ext_vector_type(8)));

constexpr int   T_LEN = 240000;
constexpr int   BATCH = 64;
constexpr int   CHUNK = 480;          // L
constexpr int   NCHNK = 500;          // C,  L*C = T
constexpr float FB    = 0.7f;
constexpr float SR    = 48000.0f;

// ---------------------------------------------------------------- coefficients
__global__ void phaser_coeff_kernel(float* __restrict__ p) {
    int t = blockIdx.x * 256 + threadIdx.x;
    if (t >= T_LEN) return;
    float tt   = (float)t / SR;
    float frac = fmodf(0.5f * tt, 1.0f);          // phase/(2pi) = F0*t
    float tri  = (frac < 0.5f) ? (4.0f * frac - 1.0f) : (3.0f - 4.0f * frac);
    float dmin = 1000.0f * 2.0f / SR;
    float dmax = 4000.0f * 2.0f / SR;
    float lfo  = dmin + (dmax - dmin) * 0.5f * (1.0f + tri);
    float tn   = tanf(lfo);
    p[t] = (1.0f - tn) / (1.0f + tn);
}

// -------------------------------------------------- one recurrence step (5-dim)
__device__ __forceinline__ void step5(float p, float xn,
                                      float& h1, float& x1, float& h2,
                                      float& h3, float& h4) {
    float x1n = xn + FB * h4;
    float h1n = p * (x1n + h1) - x1;
    float h2n = p * (h1n + h2) - h1;
    float h3n = p * (h2n + h3) - h2;
    float h4n = p * (h3n + h4) - h3;
    h1 = h1n; x1 = x1n; h2 = h2n; h3 = h3n; h4 = h4n;
}

// ------------------------------------------------------------------ pass 1
// block = 96 threads (3 waves): tid 0..63 batch columns (zero init, real x),
// tid 64..71 basis columns e_j (j<5 unit state, j>=5 zero -> zero pad columns).
__global__ void phaser_pass1_kernel(const float* __restrict__ x,
                                    const float* __restrict__ pco,
                                    float* __restrict__ PM,   // [C][8*8] row-major
                                    float* __restrict__ Z) {  // [C][64][8] batch-major
    __shared__ float sp[CHUNK];
    const int i   = blockIdx.x;
    const int tid = threadIdx.x;
    const int t0  = i * CHUNK;
    for (int j = tid; j < CHUNK; j += 96) sp[j] = pco[t0 + j];
    __syncthreads();

    const bool isBatch = (tid < BATCH);
    const bool isBasis = (tid >= 64 && tid < 72);
    const int  j       = tid - 64;

    float h1 = 0.f, x1 = 0.f, h2 = 0.f, h3 = 0.f, h4 = 0.f;
    if (isBasis) { h1 = (j == 0); x1 = (j == 1); h2 = (j == 2); h3 = (j == 3); h4 = (j == 4); }

    // clamped row -> loads are always in-bounds and can issue unconditionally
    const float* xrow = x + (size_t)(isBatch ? tid : 0) * T_LEN + t0;
    for (int k0 = 0; k0 < CHUNK; k0 += 16) {
        __builtin_prefetch(xrow + k0 + 16, 0, 0);   // global_prefetch_b8
#pragma unroll
        for (int k = k0; k < k0 + 16; ++k) {
            float p  = sp[k];
            float xv = xrow[k];
            float xn = isBatch ? xv : 0.0f;
            step5(p, xn, h1, x1, h2, h3, h4);
        }
    }

    if (isBatch) {
        float* z = Z + (size_t)i * 512 + tid * 8;   // contiguous column
        *(float4*)(z)     = make_float4(h1, x1, h2, h3);
        *(float4*)(z + 4) = make_float4(h4, 0.f, 0.f, 0.f);
    } else if (isBasis) {
        float s[8] = { h1, x1, h2, h3, h4, 0.f, 0.f, 0.f };
        float* P = PM + (size_t)i * 64;             // column j = chunk(e_j)
#pragma unroll
        for (int r = 0; r < 8; ++r) P[r * 8 + j] = s[r];
    }
}

// ------------------------------------------------------------------ pass 2
// Single block, 4 waves x 32 lanes. Wave w owns batch columns [16w,16w+16).
// S in C/D accumulator layout: VGPR r = row r (lanes 0-15) / row r+8 (lanes 16-31).
// Per chunk: S <- P_i*S + Z_i via two chained V_WMMA_F32_16X16X4_F32.
// All loads/stores unconditional (clamped addresses + v_cndmask masking).
__global__ void phaser_pass2_kernel(const float* __restrict__ PM,
                                    const float* __restrict__ Z,
                                    float* __restrict__ SI) { // [C][64][16]
    const int lane = threadIdx.x & 31;
    const int wave = threadIdx.x >> 5;
    const int sub  = lane & 15;
    const int col  = wave * 16 + sub;      // batch column for this lane
    const bool lo  = lane < 16;
    const bool rv  = sub < 8;
    const int  prc = (rv ? sub : 0) * 8;   // clamped P row base

    v8f S = { 0.f, 0.f, 0.f, 0.f, 0.f, 0.f, 0.f, 0.f };

    for (int i = 0; i < NCHNK; ++i) {
        const float* P  = PM + (size_t)i * 64;
        const float* zc = Z  + (size_t)i * 512 + col * 8;
        float*       sc = SI + (size_t)i * 1024 + col * 16 + (lo ? 0 : 8);

        // record initial state of chunk i: every lane stores its 8 rows
        *(float4*)(sc)     = make_float4(S[0], S[1], S[2], S[3]);
        *(float4*)(sc + 4) = make_float4(S[4], S[5], S[6], S[7]);

        // A operand (16x4 f32): lanes 0-15 -> K=0/1, lanes 16-31 -> K=2/3.
        float2 pa = *(const float2*)(P + prc + (lo ? 0 : 2));
        float2 pb = *(const float2*)(P + prc + (lo ? 4 : 6));
        v2f a0, a1;
        a0.x = rv ? pa.x : 0.f;  a0.y = rv ? pa.y : 0.f;
        a1.x = rv ? pb.x : 0.f;  a1.y = rv ? pb.y : 0.f;

        // C operand: Z column (contiguous), rows 8..15 (hi lanes) are zero
        float4 z0 = *(const float4*)(zc);
        float4 z1 = *(const float4*)(zc + 4);
        v8f c;
        c[0] = lo ? z0.x : 0.f;  c[1] = lo ? z0.y : 0.f;
        c[2] = lo ? z0.z : 0.f;  c[3] = lo ? z0.w : 0.f;
        c[4] = lo ? z1.x : 0.f;  c[5] = lo ? z1.y : 0.f;
        c[6] = lo ? z1.z : 0.f;  c[7] = lo ? z1.w : 0.f;

        // B operands (4x16): lanes 0-15 rows K,K+1; lanes 16-31 rows K+2,K+3
        float s2 = __shfl(S[2], sub, 32);
        float s3 = __shfl(S[3], sub, 32);
        float s6 = __shfl(S[6], sub, 32);
        float s7 = __shfl(S[7], sub, 32);
        v2f b0, b1;
        b0.x = lo ? S[0] : s2;   b0.y = lo ? S[1] : s3;
        b1.x = lo ? S[4] : s6;   b1.y = lo ? S[5] : s7;

        v8f d = __builtin_amdgcn_wmma_f32_16x16x4_f32(false, a0, false, b0,
                                                      (short)0, c, false, false);
        d     = __builtin_amdgcn_wmma_f32_16x16x4_f32(false, a1, false, b1,
                                                      (short)0, d, false, false);
        S = d;
    }
}

// ------------------------------------------------------------------ pass 3
// block = 64 threads (2 waves), one chunk per block: replay with correct init.
__global__ void phaser_pass3_kernel(const float* __restrict__ x,
                                    const float* __restrict__ pco,
                                    const float* __restrict__ SI,
                                    float* __restrict__ out) {
    __shared__ float sp[CHUNK];
    const int i  = blockIdx.x;
    const int b  = threadIdx.x;
    const int t0 = i * CHUNK;
    for (int j = b; j < CHUNK; j += 64) sp[j] = pco[t0 + j];
    __syncthreads();

    const float* si = SI + (size_t)i * 1024 + b * 16;  // this batch's init state
    float4 s0 = *(const float4*)(si);
    float h1 = s0.x, x1 = s0.y, h2 = s0.z, h3 = s0.w;
    float h4 = si[4];

    const float* xr = x   + (size_t)b * T_LEN + t0;
    float*       yr = out + (size_t)b * T_LEN + t0;

    for (int k0 = 0; k0 < CHUNK; k0 += 16) {
        __builtin_prefetch(xr + k0 + 16, 0, 0);
#pragma unroll
        for (int k = k0; k < k0 + 16; ++k) {
            float p  = sp[k];
            float xn = xr[k];
            step5(p, xn, h1, x1, h2, h3, h4);
            yr[k] = h4 + xn;   // WET_MIX = 1
        }
    }
}

// ------------------------------------------------------------------ launcher
extern "C" void kernel_launch(void* const* d_in, const int* in_sizes, int n_in,
                              void* d_out, int out_size, void* d_ws, size_t ws_size,
                              hipStream_t stream) {
    const float* x   = (const float*)d_in[0];
    float*       out = (float*)d_out;

    // workspace layout (floats): p[T] | PM[C*64] | Z[C*512] | SI[C*1024]
    float* w  = (float*)d_ws;
    float* p  = w;
    float* PM = p  + T_LEN;
    float* Z  = PM + (size_t)NCHNK * 64;
    float* SI = Z  + (size_t)NCHNK * 512;
    // total: 240000 + 32000 + 256000 + 512000 floats ~= 4.2 MB

    phaser_coeff_kernel<<<(T_LEN + 255) / 256, 256, 0, stream>>>(p);
    phaser_pass1_kernel<<<NCHNK, 96, 0, stream>>>(x, p, PM, Z);
    phaser_pass2_kernel<<<1, 128, 0, stream>>>(PM, Z, SI);
    phaser_pass3_kernel<<<NCHNK, 64, 0, stream>>>(x, p, SI, out);
}